// GPT2Attention_90666759618616
// MI455X (gfx1250) — compile-verified
//
#include <hip/hip_runtime.h>
#include <hip/hip_bf16.h>
#include <math.h>

// ---------------------------------------------------------------------------
// Problem constants (match reference)
// ---------------------------------------------------------------------------
#define NUM_SEQS    32
#define EMBED       1024
#define NUM_HEADS   16
#define HEAD_DIM    64
#define BLOCK_SZ    16
#define MAX_CTX     4096
#define MAX_BLOCKS  256
#define QK_SCALE    0.125f   // HEAD_DIM^-0.5

// flash-decode partitioning
#define NPART       4
#define PART_TOKENS 1024

typedef __attribute__((ext_vector_type(2))) float v2f;
typedef __attribute__((ext_vector_type(8))) float v8f;

#if __has_builtin(__builtin_amdgcn_global_load_async_to_lds_b32)
#define HAVE_ASYNC_LDS 1
typedef __attribute__((address_space(1))) int GlobalInt;   // builtin wants int* (per hipcc diag)
typedef __attribute__((address_space(3))) int LocalInt;
#endif

// ---------------------------------------------------------------------------
// Kernel 1/4: fp32 GEMM via V_WMMA_F32_16X16X4_F32.
// C[32, N] = A[32, K] @ B[K, N] + bias[N]
// One wave (32 threads) computes one 16x16 output tile, stepping K by 4.
// A-frag (16x4 f32): lanes 0-15 hold M=lane, K=0..1; lanes 16-31 hold
//                    M=lane-16, K=2..3  (ISA 7.12.2 "32-bit A-Matrix 16x4")
// B-frag (4x16 f32): symmetric layout over N.
// C/D (16x16 f32):  VGPR r, lanes 0-15 -> M=r, N=lane; lanes 16-31 -> M=r+8.
// ---------------------------------------------------------------------------
__global__ __launch_bounds__(32) void gemm32_wmma_f32(
    const float* __restrict__ A,     // [32, K]
    const float* __restrict__ B,     // [K, N]
    const float* __restrict__ bias,  // [N]
    float* __restrict__ C,           // [32, N]
    int K, int N)
{
    const int tile_m = blockIdx.x;           // 0..1  (32 rows)
    const int tile_n = blockIdx.y;           // N/16 tiles
    const int lane   = threadIdx.x;          // 0..31
    const int half   = lane >> 4;            // 0: K=0,1   1: K=2,3
    const int l15    = lane & 15;

    const int arow = tile_m * 16 + l15;      // A row this lane supplies
    const int bcol = tile_n * 16 + l15;      // B col this lane supplies

    v8f acc = {};
    const float* Arow = A + (size_t)arow * K;
    for (int k = 0; k < K; k += 4) {
        const int ka = k + 2 * half;
        // A fragment: two consecutive K values of this lane's row
        v2f a = *reinterpret_cast<const v2f*>(Arow + ka);
        // B fragment: same two K rows, this lane's column
        v2f b;
        b.x = B[(size_t)(ka    ) * N + bcol];
        b.y = B[(size_t)(ka + 1) * N + bcol];
        acc = __builtin_amdgcn_wmma_f32_16x16x4_f32(
            /*neg_a=*/false, a, /*neg_b=*/false, b,
            /*c_mod=*/(short)0, acc, /*reuse_a=*/false, /*reuse_b=*/false);
    }

    const float bv = bias[bcol];
#pragma unroll
    for (int r = 0; r < 8; ++r) {
        const int row = tile_m * 16 + r + 8 * half;
        C[(size_t)row * N + tile_n * 16 + l15] = acc[r] + bv;
    }
}

// ---------------------------------------------------------------------------
// Kernel 2/4: flash-decode partial attention.
// Grid (seq, head, partition). Each workgroup handles up to PART_TOKENS
// tokens of the context and emits UNNORMALIZED partials:
//   part_m = chunk max score, part_l = sum exp(score - m),
//   part_acc[64] = sum exp(score - m) * V
// Bandwidth-bound: coalesced 256B/token K/V streams + global_prefetch_b8.
// The new token's k/v come from the qkv buffer (inputs must not be mutated).
// ---------------------------------------------------------------------------
__global__ __launch_bounds__(256) void paged_attn_partial(
    const float* __restrict__ qkv,          // [32, 3*EMBED]
    const float* __restrict__ key_cache,    // [NB, 16, 16, 64]
    const float* __restrict__ value_cache,  // [NB, 16, 16, 64]
    const int*   __restrict__ block_tables, // [32, 256]
    const int*   __restrict__ context_lens, // [32]
    float* __restrict__ part_m,             // [32, 16, NPART]
    float* __restrict__ part_l,             // [32, 16, NPART]
    float* __restrict__ part_acc)           // [32, 16, NPART, 64]
{
    __shared__ __align__(16) float s_scores[PART_TOKENS];
    __shared__ __align__(16) float s_q[HEAD_DIM];
    __shared__ float s_red[8];
    __shared__ float s_accs[4][HEAD_DIM];

    const int s   = blockIdx.x;
    const int h   = blockIdx.y;
    const int p   = blockIdx.z;
    const int tid = threadIdx.x;
    const int ctx_len = context_lens[s];
    const int last    = ctx_len - 1;
    const int t0      = p * PART_TOKENS;
    const int t1      = (ctx_len < t0 + PART_TOKENS) ? ctx_len : (t0 + PART_TOKENS);
    const int pidx    = (s * NUM_HEADS + h) * NPART + p;

    // Uniform early-out for empty partitions: write identity partials.
    if (t0 >= ctx_len) {
        if (tid == 0) { part_m[pidx] = -INFINITY; part_l[pidx] = 0.f; }
        if (tid < HEAD_DIM) part_acc[(size_t)pidx * HEAD_DIM + tid] = 0.f;
        return;
    }

    const float* qkv_s = qkv + (size_t)s * 3 * EMBED;

    // Stage q into LDS (async-to-LDS path when the toolchain exposes it).
#if defined(HAVE_ASYNC_LDS)
    if (tid < HEAD_DIM) {
        __builtin_amdgcn_global_load_async_to_lds_b32(
            (GlobalInt*)(qkv_s + h * HEAD_DIM + tid),
            (LocalInt*)(s_q + tid), 0, 0);
    }
#if __has_builtin(__builtin_amdgcn_s_wait_asynccnt)
    __builtin_amdgcn_s_wait_asynccnt(0);
#else
    asm volatile("s_wait_asynccnt 0" ::: "memory");
#endif
#else
    if (tid < HEAD_DIM) s_q[tid] = qkv_s[h * HEAD_DIM + tid];
#endif
    __syncthreads();

    // ---- phase 1: scores = (K @ q) * scale, running max ----
    float lmax = -INFINITY;
    for (int t = t0 + tid; t < t1; t += 256) {
        const float4* kp;
        if (t == last) {
            kp = reinterpret_cast<const float4*>(qkv_s + EMBED + h * HEAD_DIM);
        } else {
            const int blk = block_tables[s * MAX_BLOCKS + (t >> 4)];
            kp = reinterpret_cast<const float4*>(
                key_cache + (((size_t)blk * BLOCK_SZ + (t & 15)) * NUM_HEADS + h) * HEAD_DIM);
        }
        // prefetch the K vector this thread needs next iteration
        const int tn = t + 256;
        if (tn < t1) {
            const int blk2 = block_tables[s * MAX_BLOCKS + (tn >> 4)];
            __builtin_prefetch(
                key_cache + (((size_t)blk2 * BLOCK_SZ + (tn & 15)) * NUM_HEADS + h) * HEAD_DIM,
                0, 0);
        }
        const float4* qp = reinterpret_cast<const float4*>(s_q);
        float dot = 0.f;
#pragma unroll
        for (int i = 0; i < HEAD_DIM / 4; ++i) {
            const float4 kv = kp[i];
            const float4 qv = qp[i];
            dot += kv.x * qv.x + kv.y * qv.y + kv.z * qv.z + kv.w * qv.w;
        }
        const float sc = dot * QK_SCALE;
        s_scores[t - t0] = sc;
        lmax = fmaxf(lmax, sc);
    }
    // wave32 max reduce, then cross-wave via LDS
#pragma unroll
    for (int off = 16; off > 0; off >>= 1)
        lmax = fmaxf(lmax, __shfl_xor(lmax, off, 32));
    if ((tid & 31) == 0) s_red[tid >> 5] = lmax;
    __syncthreads();
    float gmax = s_red[0];
#pragma unroll
    for (int i = 1; i < 8; ++i) gmax = fmaxf(gmax, s_red[i]);
    __syncthreads();   // s_red reused below

    // ---- phase 2: exponentiate in LDS, running (unnormalized) sum ----
    float lsum = 0.f;
    for (int t = t0 + tid; t < t1; t += 256) {
        const float e = __expf(s_scores[t - t0] - gmax);
        s_scores[t - t0] = e;
        lsum += e;
    }
#pragma unroll
    for (int off = 16; off > 0; off >>= 1)
        lsum += __shfl_xor(lsum, off, 32);
    if ((tid & 31) == 0) s_red[tid >> 5] = lsum;
    __syncthreads();
    float gsum = 0.f;
#pragma unroll
    for (int i = 0; i < 8; ++i) gsum += s_red[i];
    __syncthreads();

    // ---- phase 3: acc[d] = sum_t p_t * V[t, d]  (unnormalized) ----
    // threads (g, d): g in 0..3 strides over tokens, d in 0..63 over dims;
    // for fixed t, 64 lanes read 256B contiguous -> coalesced.
    const int d = tid & 63;
    const int g = tid >> 6;
    float acc = 0.f;
    for (int t = t0 + g; t < t1; t += 4) {
        const float pw = s_scores[t - t0];
        float vv;
        if (t == last) {
            vv = qkv_s[2 * EMBED + h * HEAD_DIM + d];
        } else {
            const int blk = block_tables[s * MAX_BLOCKS + (t >> 4)];
            vv = value_cache[(((size_t)blk * BLOCK_SZ + (t & 15)) * NUM_HEADS + h) * HEAD_DIM + d];
        }
        acc += pw * vv;
    }
    s_accs[g][d] = acc;
    __syncthreads();
    if (g == 0) {
        part_acc[(size_t)pidx * HEAD_DIM + d] =
            s_accs[0][d] + s_accs[1][d] + s_accs[2][d] + s_accs[3][d];
        if (d == 0) { part_m[pidx] = gmax; part_l[pidx] = gsum; }
    }
}

// ---------------------------------------------------------------------------
// Kernel 3/4: combine partials with log-sum-exp merge.
// Grid (seq, head), 64 threads (one per dim).
// ---------------------------------------------------------------------------
__global__ __launch_bounds__(64) void paged_attn_reduce(
    const float* __restrict__ part_m,
    const float* __restrict__ part_l,
    const float* __restrict__ part_acc,
    float* __restrict__ ctx_out)            // [32, EMBED]
{
    const int s = blockIdx.x;
    const int h = blockIdx.y;
    const int d = threadIdx.x;
    const int base = (s * NUM_HEADS + h) * NPART;

    float ms[NPART];
    float gm = -INFINITY;
#pragma unroll
    for (int p = 0; p < NPART; ++p) {
        ms[p] = part_m[base + p];
        gm = fmaxf(gm, ms[p]);
    }
    float l = 0.f, val = 0.f;
#pragma unroll
    for (int p = 0; p < NPART; ++p) {
        const float w = __expf(ms[p] - gm);     // 0 for empty partitions
        l   += w * part_l[base + p];
        val += w * part_acc[(size_t)(base + p) * HEAD_DIM + d];
    }
    ctx_out[(size_t)s * EMBED + h * HEAD_DIM + d] = val / l;
}

// ---------------------------------------------------------------------------
// Host launcher
// ---------------------------------------------------------------------------
extern "C" void kernel_launch(void* const* d_in, const int* in_sizes, int n_in,
                              void* d_out, int out_size, void* d_ws, size_t ws_size,
                              hipStream_t stream) {
    const float* hidden   = (const float*)d_in[0];
    const float* w_attn   = (const float*)d_in[1];
    const float* b_attn   = (const float*)d_in[2];
    const float* w_proj   = (const float*)d_in[3];
    const float* b_proj   = (const float*)d_in[4];
    const float* k_cache  = (const float*)d_in[5];
    const float* v_cache  = (const float*)d_in[6];
    const int*   btables  = (const int*)d_in[7];
    const int*   ctx_lens = (const int*)d_in[8];
    float* out = (float*)d_out;

    // workspace layout (floats)
    float* qkv      = (float*)d_ws;                                  // 32*3072
    float* ctxb     = qkv      + (size_t)NUM_SEQS * 3 * EMBED;       // 32*1024
    float* part_m   = ctxb     + (size_t)NUM_SEQS * EMBED;           // 32*16*NPART
    float* part_l   = part_m   + (size_t)NUM_SEQS * NUM_HEADS * NPART;
    float* part_acc = part_l   + (size_t)NUM_SEQS * NUM_HEADS * NPART;
                                     // 32*16*NPART*64

    // QKV projection: 2 x 192 tiles of 16x16, one wave each
    {
        dim3 grid(2, (3 * EMBED) / 16);
        gemm32_wmma_f32<<<grid, 32, 0, stream>>>(hidden, w_attn, b_attn, qkv,
                                                 EMBED, 3 * EMBED);
    }
    // Flash-decode paged attention: partials then reduce
    {
        dim3 grid(NUM_SEQS, NUM_HEADS, NPART);
        paged_attn_partial<<<grid, 256, 0, stream>>>(qkv, k_cache, v_cache,
                                                     btables, ctx_lens,
                                                     part_m, part_l, part_acc);
    }
    {
        dim3 grid(NUM_SEQS, NUM_HEADS);
        paged_attn_reduce<<<grid, 64, 0, stream>>>(part_m, part_l, part_acc,
                                                   ctxb);
    }
    // Output projection: 2 x 64 tiles
    {
        dim3 grid(2, EMBED / 16);
        gemm32_wmma_f32<<<grid, 32, 0, stream>>>(ctxb, w_proj, b_proj, out,
                                                 EMBED, EMBED);
    }
}